// VectorQuantizer1d_35304631174227
// MI455X (gfx1250) — compile-verified
//
#include <hip/hip_runtime.h>

typedef __attribute__((ext_vector_type(2))) float v2f;
typedef __attribute__((ext_vector_type(8))) float v8f;

#define LATENT     1024
#define WORD       64
#define NB_WORDS   8192
#define BATCH      2048
#define NROWS      (BATCH * (LATENT / WORD))   // 32768 z-vectors of length 64
#define KSTEPS     (WORD / 4)                  // 16 wmma k-steps (K=4 each)
#define COLTILES   (NB_WORDS / 16)             // 512 column tiles
#define WAVES_PER_BLOCK 8
#define COMMIT_VQ  2.5f

// ---------------------------------------------------------------------------
// Kernel 0: codebook row norms ||e_j||^2 (argmin term), zero loss accumulator.
// ---------------------------------------------------------------------------
__global__ void vq_norms_kernel(const float* __restrict__ emb,
                                float* __restrict__ normE,
                                float* __restrict__ acc) {
    int j = blockIdx.x * blockDim.x + threadIdx.x;
    if (j < NB_WORDS) {
        const float* row = emb + j * WORD;
        float s = 0.f;
#pragma unroll
        for (int k = 0; k < WORD; ++k) s += row[k] * row[k];
        normE[j] = s;
    }
    if (j == 0) *acc = 0.f;
}

// ---------------------------------------------------------------------------
// Kernel 1: fused fp32-WMMA distance GEMM + argmin + codeword gather + loss.
// One wave32 owns a 16-row tile of Z and sweeps all 8192 codewords.
// ---------------------------------------------------------------------------
__global__ __launch_bounds__(256) void vq_main_kernel(
    const float* __restrict__ z,
    const float* __restrict__ emb,
    const float* __restrict__ normE,
    float* __restrict__ out,
    float* __restrict__ acc) {

    const int lane = threadIdx.x & 31;
    const int wave = threadIdx.x >> 5;
    const int l16  = lane & 15;
    const int half = lane >> 4;            // 0: lanes 0-15, 1: lanes 16-31

    const int rowTile = blockIdx.x * WAVES_PER_BLOCK + wave;  // 0..2047
    const int rowBase = rowTile * 16;

    // --- A fragments: row M = l16, K pairs {k0,k0+1} (low half) / {k0+2,k0+3}
    v2f a[KSTEPS];
    {
        const float* zrow = z + (size_t)(rowBase + l16) * WORD + half * 2;
#pragma unroll
        for (int ks = 0; ks < KSTEPS; ++ks) {
            a[ks].x = zrow[ks * 4 + 0];
            a[ks].y = zrow[ks * 4 + 1];
        }
    }

    // Per-lane running argmin. C/D layout: reg r, this lane -> row r+8*half,
    // column (coltile*16 + l16).
    float minval[8];
    int   minidx[8];
#pragma unroll
    for (int r = 0; r < 8; ++r) { minval[r] = 3.4e38f; minidx[r] = 0; }

    for (int ct = 0; ct < COLTILES; ++ct) {
        const int col0 = ct * 16;
        const float* erow = emb + (size_t)(col0 + l16) * WORD + half * 2;

        // Two independent accumulator chains to break the WMMA RAW chain.
        v8f c0 = {};
        v8f c1 = {};
#pragma unroll
        for (int ks = 0; ks < KSTEPS; ks += 2) {
            v2f b0, b1;
            b0.x = erow[ks * 4 + 0];
            b0.y = erow[ks * 4 + 1];
            b1.x = erow[ks * 4 + 4];
            b1.y = erow[ks * 4 + 5];
            c0 = __builtin_amdgcn_wmma_f32_16x16x4_f32(
                     false, a[ks],     false, b0, (short)0, c0, false, false);
            c1 = __builtin_amdgcn_wmma_f32_16x16x4_f32(
                     false, a[ks + 1], false, b1, (short)0, c1, false, false);
        }

        const int   col = col0 + l16;
        const float ne  = normE[col];
#pragma unroll
        for (int r = 0; r < 8; ++r) {
            // squared distance minus per-row constant ||z||^2
            float d = ne - 2.0f * (c0[r] + c1[r]);
            if (d < minval[r]) { minval[r] = d; minidx[r] = col; }
        }
    }

    // Reduce argmin across the 16 columns held within each half-wave.
    // xor masks 8,4,2,1 never cross the lane-16 boundary in wave32.
#pragma unroll
    for (int r = 0; r < 8; ++r) {
#pragma unroll
        for (int m = 8; m >= 1; m >>= 1) {
            float ov = __shfl_xor(minval[r], m, 32);
            int   oi = __shfl_xor(minidx[r], m, 32);
            if (ov < minval[r] || (ov == minval[r] && oi < minidx[r])) {
                minval[r] = ov; minidx[r] = oi;
            }
        }
    }

    // Publish the 16 row-winners (lane0 -> rows 0-7, lane16 -> rows 8-15).
    __shared__ int s_idx[WAVES_PER_BLOCK][16];
    if (l16 == 0) {
#pragma unroll
        for (int r = 0; r < 8; ++r) s_idx[wave][half * 8 + r] = minidx[r];
    }
    __syncthreads();

    // Gather z_q = emb[idx], write output (straight-through forward == z_q),
    // and accumulate sum((z_q - z)^2): 16 rows x 64 = 1024 elems, 32/lane.
    float lsum = 0.f;
#pragma unroll
    for (int e = 0; e < 32; ++e) {
        int t   = e * 32 + lane;         // 0..1023
        int r   = t >> 6;                // row within tile
        int k   = t & 63;
        int idx = s_idx[wave][r];
        size_t o = (size_t)(rowBase + r) * WORD + k;
        float q  = emb[(size_t)idx * WORD + k];
        float zv = z[o];
        out[o]   = q;
        float df = q - zv;
        lsum += df * df;
    }
#pragma unroll
    for (int m = 16; m >= 1; m >>= 1) lsum += __shfl_xor(lsum, m, 32);
    if (lane == 0) atomicAdd(acc, lsum);
}

// ---------------------------------------------------------------------------
// Kernel 2: loss = (1 + COMMIT_VQ) * sum / (BATCH*LATENT)
// ---------------------------------------------------------------------------
__global__ void vq_finish_kernel(const float* __restrict__ acc,
                                 float* __restrict__ loss_out) {
    *loss_out = (1.0f + COMMIT_VQ) * (*acc) * (1.0f / (float)(BATCH * LATENT));
}

extern "C" void kernel_launch(void* const* d_in, const int* in_sizes, int n_in,
                              void* d_out, int out_size, void* d_ws, size_t ws_size,
                              hipStream_t stream) {
    const float* z    = (const float*)d_in[0];   // z_mean (2048,1024)
    // d_in[1] = z_log_var: unused by the reference computation
    const float* emb  = (const float*)d_in[2];   // (8192,64)

    float* out   = (float*)d_out;                // z_q_st, 2048*1024 floats
    float* loss  = out + (size_t)BATCH * LATENT; // scalar loss
    float* normE = (float*)d_ws;                 // 8192 floats
    float* acc   = normE + NB_WORDS;             // 1 float

    vq_norms_kernel<<<(NB_WORDS + 255) / 256, 256, 0, stream>>>(emb, normE, acc);
    vq_main_kernel<<<NROWS / (16 * WAVES_PER_BLOCK), 256, 0, stream>>>(
        z, emb, normE, out, acc);
    vq_finish_kernel<<<1, 1, 0, stream>>>(acc, loss);
}